// DepthwiseTemporalConv_24962349924982
// MI455X (gfx1250) — compile-verified
//
#include <hip/hip_runtime.h>
#include <hip/hip_bf16.h>

typedef __attribute__((ext_vector_type(2))) float v2f;
typedef __attribute__((ext_vector_type(8))) float v8f;

#define CC   256   // channels
#define TT   64    // temporal length (= K of the conv)
#define HW   1024  // H*W columns per (b,c) slab
#define COLS_PER_WAVE 16
#define WAVES_PER_WG  8
#define COLS_PER_WG   (COLS_PER_WAVE * WAVES_PER_WG)  // 128

// y[b,c,t,hw] = sum_{s=t}^{63} w[c, s-t] * x[b,c,s,hw]
// Per (b,c): Y(64 x 1024) = M_c(64 x 64, upper-tri Toeplitz) * X(64 x 1024)
//
// Toeplitz structure => the WMMA A-tile for (mt, kk) depends only on
// r = kk - 4*mt, so only 16 distinct A tiles exist; precompute them in VGPRs.
__global__ __launch_bounds__(256)
void DepthwiseTemporalConv_kernel(const float* __restrict__ x,
                                  const float* __restrict__ w,
                                  float* __restrict__ y) {
    __shared__ float sw[TT];

    const int tid    = threadIdx.x;
    const int wave   = tid >> 5;
    const int lane   = tid & 31;
    const int lane16 = lane & 15;
    const int hi     = lane >> 4;          // 0: lanes 0-15, 1: lanes 16-31

    // blockIdx.x = bc * 8 + hwTile   (bc = b*C + c)
    const int hwTile = blockIdx.x & 7;
    const int bc     = blockIdx.x >> 3;
    const int c      = bc & (CC - 1);

    // Stage this channel's 64 weights in LDS.
    if (tid < TT) sw[tid] = w[c * TT + tid];
    __syncthreads();

    const int col = hwTile * COLS_PER_WG + wave * COLS_PER_WAVE + lane16;
    const float* __restrict__ xb = x + (size_t)bc * (TT * HW);
    float* __restrict__ yb       = y + (size_t)bc * (TT * HW);

    // ---- Preload B in WMMA 4x16 f32 lane layout (streaming, read-once). ----
    // K-step kk covers rows s = 4kk..4kk+3.
    // elem .x: lanes 0-15 -> row 4kk,   lanes 16-31 -> row 4kk+2
    // elem .y: lanes 0-15 -> row 4kk+1, lanes 16-31 -> row 4kk+3
    v2f Bv[16];
#pragma unroll
    for (int kk = 0; kk < 16; ++kk) {
        const int s0 = 4 * kk + 2 * hi;
        v2f b;
        b.x = __builtin_nontemporal_load(&xb[(size_t)s0 * HW + col]);
        b.y = __builtin_nontemporal_load(&xb[(size_t)(s0 + 1) * HW + col]);
        Bv[kk] = b;
    }

    // ---- Precompute the 16 distinct A tiles (r = kk - 4*mt). ----
    // A 16x4 f32 lane layout: elem j holds A[m=lane16, k=2*hi+j],
    // A[t, s] = w[s - t] for s >= t else 0; d0 = 4r + 2hi - lane16.
    v2f Areg[16];
#pragma unroll
    for (int r = 0; r < 16; ++r) {
        const int d0 = 4 * r + 2 * hi - lane16;
        v2f a;
        if (r >= 4) {                      // compile-time: d0 >= 1 always
            a.x = sw[d0];
            a.y = sw[d0 + 1];
        } else {                           // near-diagonal: mask below diag
            const int i0 = d0 < 0 ? 0 : d0;
            const int i1 = (d0 + 1) < 0 ? 0 : (d0 + 1);
            const float l0 = sw[i0];
            const float l1 = sw[i1];
            a.x = (d0     >= 0) ? l0 : 0.f;
            a.y = (d0 + 1 >= 0) ? l1 : 0.f;
        }
        Areg[r] = a;
    }

    // ---- 40 register-only WMMAs; 4 interleaved accumulator chains. ----
    v8f acc[4];
#pragma unroll
    for (int mt = 0; mt < 4; ++mt)
        acc[mt] = (v8f){0.f, 0.f, 0.f, 0.f, 0.f, 0.f, 0.f, 0.f};

#pragma unroll
    for (int kk = 0; kk < 16; ++kk) {
#pragma unroll
        for (int mt = 0; mt < 4; ++mt) {
            if (kk >= 4 * mt) {            // skip below-diagonal K-steps
                acc[mt] = __builtin_amdgcn_wmma_f32_16x16x4_f32(
                    false, Areg[kk - 4 * mt], false, Bv[kk],
                    (short)0, acc[mt], false, false);
            }
        }
    }

    // ---- Store D (streaming). VGPR v -> row M = v + 8*hi, col N = lane16. ----
#pragma unroll
    for (int mt = 0; mt < 4; ++mt) {
#pragma unroll
        for (int v = 0; v < 8; ++v) {
            const int row = mt * 16 + v + 8 * hi;
            __builtin_nontemporal_store(acc[mt][v], &yb[(size_t)row * HW + col]);
        }
    }
}

extern "C" void kernel_launch(void* const* d_in, const int* in_sizes, int n_in,
                              void* d_out, int out_size, void* d_ws, size_t ws_size,
                              hipStream_t stream) {
    (void)in_sizes; (void)n_in; (void)out_size; (void)d_ws; (void)ws_size;
    const float* x = (const float*)d_in[0];   // (B, C, T, H, W) fp32
    const float* w = (const float*)d_in[1];   // (C, 1, K)       fp32
    float* y = (float*)d_out;                 // (B, C, T, H, W) fp32

    const int B = 4;
    const int blocks = B * CC * (HW / COLS_PER_WG);  // 4*256*8 = 8192
    DepthwiseTemporalConv_kernel<<<blocks, 256, 0, stream>>>(x, w, y);
}